// LinkPredict_53068615909712
// MI455X (gfx1250) — compile-verified
//
#include <hip/hip_runtime.h>
#include <hip/hip_bf16.h>

// ---------------------------------------------------------------------------
// RGCN link-predict for MI455X (gfx1250, wave32).
//
// Basis-decomposition reassociation:
//   msg_e = norm_e * sum_b coeff[etype_e,b] * (h[src_e] @ basis_b)
// -> dense part is 8 GEMMs [N,128]x[128,128] per layer (B=8), not R=32.
// HB[b] = h @ basis_b (123 MB) is staged L2-resident (192 MB L2), so the
// random per-edge gathers and segment-sum atomics are served from L2.
//
// GEMM: V_WMMA_F32_16X16X4_F32, register-blocked 5x4 tiles per wave
// (80x64 output / wave): 9 b64 loads feed 20 WMMAs per K-step.
// basis is pre-packed so the per-lane (k,k+1) operand pair is one b64 load.
// ---------------------------------------------------------------------------

#define NN   30000       // nodes
#define DD   128         // feature dim
#define NB   8           // bases
#define NR   32          // relations
#define RT   5           // 16-row tiles per wave  (1875 row tiles = 375 * 5)
#define CT   4           // 16-col tiles per wave  (8 col tiles = 2 * 4)

typedef __attribute__((ext_vector_type(2))) float v2f;
typedef __attribute__((ext_vector_type(8))) float v8f;

// Repack basis [LB, 128, 128] -> Bp pairs: Bp[bb][k2][col] = {B[2k2,col], B[2k2+1,col]}
// Runs once; 1 MB total.
__global__ void __launch_bounds__(256)
pack_basis(const float* __restrict__ basis, float* __restrict__ Bp, int total_pairs)
{
    const int i = blockIdx.x * blockDim.x + threadIdx.x;
    if (i < total_pairs) {
        const int col = i & (DD - 1);
        const int k2  = (i >> 7) & 63;
        const int bb  = i >> 13;                  // combined (layer*NB + b)
        const float* s = basis + (size_t)bb * DD * DD + (size_t)(2 * k2) * DD + col;
        v2f p; p.x = s[0]; p.y = s[DD];
        ((v2f*)Bp)[i] = p;
    }
}

// One wave computes an 80x64 block of HB[b] = h @ basis_b.
// grid = (375, 2, NB), block = 32 (one wave).
// A (16x4 f32): lane holds A[m, k0+{0,1}], m=lane%16, k-half by lane/16.
// B (4x16 f32): lane holds B[k0+{0,1}, n], n=lane%16 -> one b64 from packed Bp.
// C/D (16x16 f32): VGPR j holds row j (lanes 0-15) / row j+8 (lanes 16-31).
__global__ void __launch_bounds__(32)
rgcn_gemm_hb(const float* __restrict__ h,      // [NN, DD]
             const float* __restrict__ Bp,     // [NB, 64, DD] float2 (layer slice)
             float* __restrict__ HB)           // [NB, NN, DD]
{
    const int lane = threadIdx.x;
    const int nblk = blockIdx.x;   // 80-row block
    const int kblk = blockIdx.y;   // 64-col block
    const int b    = blockIdx.z;   // basis index

    const float* A  = h + (size_t)nblk * (RT * 16) * DD;
    const v2f*   Bq = (const v2f*)(Bp + (size_t)b * DD * DD);
    float*       D  = HB + (size_t)b * NN * DD + (size_t)nblk * (RT * 16) * DD
                         + (size_t)kblk * 64;

    const int m  = lane & 15;          // A row within tile
    const int kh = (lane >> 4) * 2;    // K sub-offset (0 or 2)
    const int nc = lane & 15;          // B/D column within tile

    const v8f zero = {0.f, 0.f, 0.f, 0.f, 0.f, 0.f, 0.f, 0.f};
    v8f c[RT][CT];
    #pragma unroll
    for (int rt = 0; rt < RT; ++rt)
        #pragma unroll
        for (int ct = 0; ct < CT; ++ct)
            c[rt][ct] = zero;

    for (int kk = 0; kk < DD; kk += 4) {
        const int k2 = (kk >> 1) + (lane >> 4);    // packed-pair row

        v2f a[RT], bb[CT];
        #pragma unroll
        for (int rt = 0; rt < RT; ++rt)
            a[rt] = *(const v2f*)(A + (size_t)(rt * 16 + m) * DD + kk + kh);
        #pragma unroll
        for (int ct = 0; ct < CT; ++ct)
            bb[ct] = Bq[(size_t)k2 * DD + kblk * 64 + ct * 16 + nc];

        #pragma unroll
        for (int rt = 0; rt < RT; ++rt)
            #pragma unroll
            for (int ct = 0; ct < CT; ++ct)
                c[rt][ct] = __builtin_amdgcn_wmma_f32_16x16x4_f32(
                    /*neg_a=*/false, a[rt], /*neg_b=*/false, bb[ct],
                    /*c_mod=*/(short)0, c[rt][ct],
                    /*reuse_a=*/false, /*reuse_b=*/false);
    }

    const int rbase = (lane >> 4) * 8;
    #pragma unroll
    for (int rt = 0; rt < RT; ++rt)
        #pragma unroll
        for (int ct = 0; ct < CT; ++ct)
            #pragma unroll
            for (int j = 0; j < 8; ++j)
                D[(size_t)(rt * 16 + rbase + j) * DD + ct * 16 + nc] = c[rt][ct][j];
}

// One 128-thread block per edge; lane = output channel k.
// acc = sum_b coeff[r,b] * HB[b, src, k];  agg[dst,k] += norm * acc
// HB reads and agg atomics are L2-resident by construction.
__global__ void __launch_bounds__(128)
rgcn_edge(const float* __restrict__ HB,     // [NB, NN, DD]
          const int*   __restrict__ src,
          const int*   __restrict__ dst,
          const int*   __restrict__ etype,
          const float* __restrict__ norm,   // [E,1]
          const float* __restrict__ coeff,  // [NR, NB] (layer slice)
          float* __restrict__ agg)          // [NN, DD]
{
    const int e = blockIdx.x;
    const int k = threadIdx.x;

    const int   s = src[e];
    const int   d = dst[e];
    const int   r = etype[e];
    const float w = norm[e];

    const float* cf = coeff + (size_t)r * NB;
    const float* hb = HB + (size_t)s * DD + k;

    float acc = 0.f;
    #pragma unroll
    for (int b = 0; b < NB; ++b)
        acc = fmaf(cf[b], hb[(size_t)b * NN * DD], acc);

    atomicAdd(agg + (size_t)d * DD + k, w * acc);
}

// h_next = relu(agg + bias)
__global__ void __launch_bounds__(256)
rgcn_bias_relu(const float* __restrict__ agg,
               const float* __restrict__ bias,   // [DD] (layer slice)
               float* __restrict__ out, int total)
{
    const int i = blockIdx.x * blockDim.x + threadIdx.x;
    if (i < total) {
        const float v = agg[i] + bias[i & (DD - 1)];
        out[i] = v > 0.f ? v : 0.f;
    }
}

// out[i,:] = tab[idx[i],:]   (128-wide rows)
__global__ void __launch_bounds__(128)
gather_rows(const float* __restrict__ tab,
            const int*   __restrict__ idx,
            float* __restrict__ out)
{
    const int i = blockIdx.x;
    const int k = threadIdx.x;
    out[(size_t)i * DD + k] = tab[(size_t)idx[i] * DD + k];
}

extern "C" void kernel_launch(void* const* d_in, const int* in_sizes, int n_in,
                              void* d_out, int out_size, void* d_ws, size_t ws_size,
                              hipStream_t stream) {
    const float* p_feats = (const float*)d_in[0];
    const float* n_feats = (const float*)d_in[1];
    const float* w_rel   = (const float*)d_in[2];   // [16, 128]
    const float* coeff   = (const float*)d_in[3];   // [2, 32, 8]
    const float* basis   = (const float*)d_in[4];   // [2, 8, 128, 128]
    const float* bias    = (const float*)d_in[5];   // [2, 128]
    const int*   p_src   = (const int*)d_in[6];
    const int*   p_dst   = (const int*)d_in[7];
    const int*   p_et    = (const int*)d_in[8];
    const float* p_norm  = (const float*)d_in[9];
    const int*   n_src   = (const int*)d_in[10];
    const int*   n_dst   = (const int*)d_in[11];
    const int*   n_et    = (const int*)d_in[12];
    const float* n_norm  = (const float*)d_in[13];
    const int*   pg_head = (const int*)d_in[14];
    const int*   pg_tail = (const int*)d_in[15];
    const int*   pg_et   = (const int*)d_in[16];
    const int*   ng_head = (const int*)d_in[17];
    const int*   ng_tail = (const int*)d_in[18];

    const int E = in_sizes[6];    // 480000
    const int P = in_sizes[14];   // 100000

    // Workspace: HB (123 MB) | agg (15.4 MB) | h_tmp (15.4 MB) | Bp (1 MB)
    float* HB   = (float*)d_ws;
    float* agg  = HB   + (size_t)NB * NN * DD;
    float* htmp = agg  + (size_t)NN * DD;
    float* Bp   = htmp + (size_t)NN * DD;          // [2, NB, 64, DD] float2

    // Output layout (reference tuple order, concatenated flat).
    float* out    = (float*)d_out;
    float* p_head = out;
    float* p_tail = out + (size_t)1 * P * DD;
    float* r_emb  = out + (size_t)2 * P * DD;
    float* n_head = out + (size_t)3 * P * DD;
    float* n_tail = out + (size_t)4 * P * DD;
    float* p_h    = out + (size_t)5 * P * DD;
    float* n_h    = p_h + (size_t)NN * DD;

    // One-shot basis repack (both layers): 2*NB*64*128 float2 pairs.
    const int total_pairs = 2 * NB * 64 * DD;
    pack_basis<<<(total_pairs + 255) / 256, 256, 0, stream>>>(basis, Bp, total_pairs);

    const dim3 ggrid(NN / (RT * 16), DD / (CT * 16), NB), gblk(32);
    const int  total = NN * DD;

    auto encode = [&](const float* feats, const int* src, const int* dst,
                      const int* et, const float* nm, float* h_out) {
        const float* h = feats;
        for (int l = 0; l < 2; ++l) {
            rgcn_gemm_hb<<<ggrid, gblk, 0, stream>>>(
                h, Bp + (size_t)l * NB * DD * DD, HB);
            hipMemsetAsync(agg, 0, (size_t)total * sizeof(float), stream);
            rgcn_edge<<<dim3(E), dim3(DD), 0, stream>>>(
                HB, src, dst, et, nm, coeff + (size_t)l * NR * NB, agg);
            float* dest = (l == 0) ? htmp : h_out;
            rgcn_bias_relu<<<(total + 255) / 256, 256, 0, stream>>>(
                agg, bias + (size_t)l * DD, dest, total);
            h = dest;
        }
    };

    encode(p_feats, p_src, p_dst, p_et, p_norm, p_h);
    encode(n_feats, n_src, n_dst, n_et, n_norm, n_h);

    gather_rows<<<dim3(P), dim3(DD), 0, stream>>>(p_h,   pg_head, p_head);
    gather_rows<<<dim3(P), dim3(DD), 0, stream>>>(p_h,   pg_tail, p_tail);
    gather_rows<<<dim3(P), dim3(DD), 0, stream>>>(w_rel, pg_et,   r_emb);
    gather_rows<<<dim3(P), dim3(DD), 0, stream>>>(n_h,   ng_head, n_head);
    gather_rows<<<dim3(P), dim3(DD), 0, stream>>>(n_h,   ng_tail, n_tail);
}